// MMRouting_18116172054896
// MI455X (gfx1250) — compile-verified
//
#include <hip/hip_runtime.h>

typedef float v2f __attribute__((ext_vector_type(2)));
typedef float v4f __attribute__((ext_vector_type(4)));
typedef float v8f __attribute__((ext_vector_type(8)));

#define BDIM 512
#define IDIM 7
#define DDIM 512
#define JDIM 64
#define EDIM 128
#define NCOL (JDIM * EDIM)   // 8192
#define ITERS 10

// ---------------------------------------------------------------------------
// Kernel 1: fW[b][i][n] = sum_d f[b][i][d] * W[i][d][n],  n = j*128+e
// f32 WMMA 16x16x4.  Block = 256 threads (8 wave32s).
// Block tile M=128, N=128, K staged 32 at a time.
// Wave grid 4(M) x 2(N): each wave owns M=32 x N=64 -> 8 v8f accumulators,
// reusing each B fragment across 2 A strips (0.75 ds_load_b64 per WMMA).
// LDS holds k-PAIR-interleaved tiles so each WMMA operand is one ds_load_b64.
// ---------------------------------------------------------------------------
__global__ __launch_bounds__(256)
void capsule_gemm_wmma(const float* __restrict__ f,
                       const float* __restrict__ w,
                       float* __restrict__ fw) {
  __shared__ v2f As2[16][128];  // [k/2][m] = {A[k][m], A[k+1][m]}
  __shared__ v2f Bs2[16][128];  // [k/2][n] = {B[k][n], B[k+1][n]}

  const int blockN = blockIdx.x * 128;
  const int blockM = blockIdx.y * 128;
  const int i      = blockIdx.z;

  const int tid  = threadIdx.x;
  const int lane = tid & 31;
  const int wave = tid >> 5;

  const int waveM = (wave >> 1) * 32;  // 0,32,64,96
  const int waveN = (wave & 1) * 64;   // 0,64

  v8f acc[2][4] = {};                  // [A strip][N tile]

  const int mIdx  = lane & 15;         // M(A) / N(B,C) position within 16
  const int kHalf = lane >> 4;         // which k-pair of the group of 4

  // A staging: thread -> (m = tid>>1, 16 consecutive k at (tid&1)*16)
  const int am  = tid >> 1;
  const int ak0 = (tid & 1) * 16;
  const float* aRow = f + ((size_t)(blockM + am) * IDIM + i) * DDIM;

  // B staging: thread -> (k-pair = tid>>4, 8 consecutive n at (tid&15)*8)
  const int bkp = tid >> 4;
  const int bn0 = (tid & 15) * 8;

  for (int kb = 0; kb < DDIM; kb += 32) {
    // ---- stage A tile (128m x 32k), k-pair interleaved ----
#pragma unroll
    for (int q = 0; q < 4; ++q) {
      v4f av = *(const v4f*)(aRow + kb + ak0 + q * 4);
      As2[((ak0 + q * 4) >> 1) + 0][am] = v2f{av.x, av.y};
      As2[((ak0 + q * 4) >> 1) + 1][am] = v2f{av.z, av.w};
    }
    // ---- stage B tile (32k x 128n), k-pair interleaved, b128 LDS stores ----
    {
      const float* b0 =
          w + ((size_t)i * DDIM + kb + 2 * bkp) * NCOL + blockN + bn0;
      const float* b1 = b0 + NCOL;
      v4f r0  = *(const v4f*)(b0);
      v4f r0b = *(const v4f*)(b0 + 4);
      v4f r1  = *(const v4f*)(b1);
      v4f r1b = *(const v4f*)(b1 + 4);
      v4f* dstq = (v4f*)&Bs2[bkp][bn0];
      dstq[0] = v4f{r0.x,  r1.x,  r0.y,  r1.y};
      dstq[1] = v4f{r0.z,  r1.z,  r0.w,  r1.w};
      dstq[2] = v4f{r0b.x, r1b.x, r0b.y, r1b.y};
      dstq[3] = v4f{r0b.z, r1b.z, r0b.w, r1b.w};
    }
    __syncthreads();

    // ---- 8 k-steps of 4: 2 A-frag + 4 B-frag loads -> 8 WMMAs ----
#pragma unroll
    for (int ks = 0; ks < 8; ++ks) {
      const int kp2 = ks * 2 + kHalf;
      v2f a0 = As2[kp2][waveM + mIdx];
      v2f a1 = As2[kp2][waveM + 16 + mIdx];
#pragma unroll
      for (int t = 0; t < 4; ++t) {
        v2f bv = Bs2[kp2][waveN + t * 16 + mIdx];
        acc[0][t] = __builtin_amdgcn_wmma_f32_16x16x4_f32(
            false, a0, false, bv, (short)0, acc[0][t], false, false);
        acc[1][t] = __builtin_amdgcn_wmma_f32_16x16x4_f32(
            false, a1, false, bv, (short)0, acc[1][t], false, false);
      }
    }
    __syncthreads();
  }

  // ---- store C: VGPR r holds M=r (lanes 0-15) / M=r+8 (lanes 16-31) ----
  const int rHi = (lane >> 4) * 8;
#pragma unroll
  for (int s = 0; s < 2; ++s) {
#pragma unroll
    for (int t = 0; t < 4; ++t) {
#pragma unroll
      for (int r2 = 0; r2 < 8; ++r2) {
        int row = blockM + waveM + s * 16 + r2 + rHi;   // b index
        size_t off =
            ((size_t)row * IDIM + i) * NCOL + blockN + waveN + t * 16 + mIdx;
        fw[off] = acc[s][t][r2];
      }
    }
  }
}

// ---------------------------------------------------------------------------
// Kernel 2: 10 routing iterations, one workgroup per b.
// Entire fW[b] slice (229 KB) + c (32 KB) lives in LDS (CDNA5: 320 KB/WGP).
// ---------------------------------------------------------------------------
__global__ __launch_bounds__(256)
void capsule_routing(const float* __restrict__ fwG,
                     const float* __restrict__ p,
                     float* __restrict__ outC,    // [B][J][E]
                     float* __restrict__ outR) {  // [B][I][J] (last iter r)
  extern __shared__ float smem[];
  float* fw = smem;                    // 7*8192 = 57344
  float* cs = fw + IDIM * NCOL;        // 8192
  float* lg = cs + NCOL;               // 448 logits
  float* rs = lg + IDIM * JDIM;        // 448 r
  float* ps = rs + IDIM * JDIM;        // 8   p[b][i]

  const int b    = blockIdx.x;
  const int tid  = threadIdx.x;
  const int lane = tid & 31;
  const int wave = tid >> 5;

  // ---- load fW[b] slice and init c = 1 ----
  {
    const v4f* src = (const v4f*)(fwG + (size_t)b * IDIM * NCOL);
    v4f* dst = (v4f*)fw;
    for (int s = tid; s < (IDIM * NCOL) / 4; s += 256) dst[s] = src[s];
    for (int s = tid; s < NCOL; s += 256) cs[s] = 1.0f;
    if (tid < IDIM) ps[tid] = p[b * IDIM + tid];
  }
  __syncthreads();

  for (int iter = 0; iter < ITERS; ++iter) {
    // ---- logits[i][j] = sum_e fW[i][j][e] * c[j][e] ----
    // 448 (i,j) pairs over 8 waves; lanes split e, shuffle-reduce (wave32).
    for (int s = 0; s < 56; ++s) {
      const int pair = wave + 8 * s;       // i*64 + j
      const int j    = pair & 63;
      const float* fr = fw + pair * EDIM;
      const float* cr = cs + j * EDIM;
      float sum = fr[lane]      * cr[lane]
                + fr[lane + 32] * cr[lane + 32]
                + fr[lane + 64] * cr[lane + 64]
                + fr[lane + 96] * cr[lane + 96];
#pragma unroll
      for (int off = 16; off >= 1; off >>= 1)
        sum += __shfl_down(sum, off, 32);
      if (lane == 0) lg[pair] = sum;
    }
    __syncthreads();

    // ---- softmax over j (64) per i: wave i handles row i ----
    if (wave < IDIM) {
      const int i = wave;
      float v0 = lg[i * 64 + lane];
      float v1 = lg[i * 64 + 32 + lane];
      float m = fmaxf(v0, v1);
#pragma unroll
      for (int off = 16; off >= 1; off >>= 1)
        m = fmaxf(m, __shfl_xor(m, off, 32));
      float e0 = __expf(v0 - m);
      float e1 = __expf(v1 - m);
      float ssum = e0 + e1;
#pragma unroll
      for (int off = 16; off >= 1; off >>= 1)
        ssum += __shfl_xor(ssum, off, 32);
      const float inv = 1.0f / ssum;
      const float r0 = e0 * inv, r1 = e1 * inv;
      rs[i * 64 + lane]      = r0;
      rs[i * 64 + 32 + lane] = r1;
      if (iter == ITERS - 1) {
        outR[(size_t)b * IDIM * JDIM + i * 64 + lane]      = r0;
        outR[(size_t)b * IDIM * JDIM + i * 64 + 32 + lane] = r1;
      }
    }
    __syncthreads();

    // ---- c[j][e] = sum_i p[i] * r[i][j] * fW[i][j][e] ----
    for (int s = 0; s < NCOL / 256; ++s) {
      const int idx = tid + 256 * s;       // j*128 + e
      const int j   = idx >> 7;
      float sum = 0.f;
#pragma unroll
      for (int i = 0; i < IDIM; ++i)
        sum += ps[i] * rs[i * 64 + j] * fw[i * NCOL + idx];
      cs[idx] = sum;
    }
    __syncthreads();
  }

  // ---- write final c ----
  for (int s = 0; s < NCOL / 256; ++s) {
    const int idx = tid + 256 * s;
    outC[(size_t)b * NCOL + idx] = cs[idx];
  }
}

// ---------------------------------------------------------------------------
extern "C" void kernel_launch(void* const* d_in, const int* in_sizes, int n_in,
                              void* d_out, int out_size, void* d_ws,
                              size_t ws_size, hipStream_t stream) {
  const float* f = (const float*)d_in[0];   // (B, I, D)
  const float* p = (const float*)d_in[1];   // (B, I)
  const float* w = (const float*)d_in[2];   // (I, D, J, E)
  float* out   = (float*)d_out;             // c (B,J,E) ++ r (B,I,J)
  float* fwbuf = (float*)d_ws;              // needs B*I*8192*4 = 117.4 MB

  dim3 g1(NCOL / 128, BDIM / 128, IDIM);    // (64, 4, 7) blocks
  capsule_gemm_wmma<<<g1, 256, 0, stream>>>(f, w, fwbuf);

  const size_t shmem =
      (size_t)(IDIM * NCOL + NCOL + 2 * IDIM * JDIM + 8) * sizeof(float);
  capsule_routing<<<BDIM, 256, shmem, stream>>>(
      fwbuf, p, out, out + (size_t)BDIM * NCOL);
}